// ReconciliationBridge_88218628260362
// MI455X (gfx1250) — compile-verified
//
#include <hip/hip_runtime.h>

typedef __attribute__((ext_vector_type(2))) float v2f;
typedef __attribute__((ext_vector_type(8))) float v8f;

#define NN 50000
#define NE 800000
#define DF 128
#define LN_EPS 1e-5f

// ---------------------------------------------------------------------------
// Stage `rows` x 128 fp32 weights from global into LDS, pre-swizzled into
// V_WMMA_F32_16X16X4_F32 B-fragment order (one conflict-free ds_load_b64 per
// fragment per lane).  Staging uses the CDNA5 async copy path: each lane
// issues global_load_async_to_lds_b32 with a per-lane LDS destination address
// (VDST) and per-lane global source (VADDR), tracked by ASYNCcnt.
//
// B-fragment layout (CDNA5 ISA): for a 4x16 B tile, lanes 0-15 hold
// K0 (vgpr0) / K0+1 (vgpr1), lanes 16-31 hold K0+2 / K0+3; N = lane&15.
// LDS word for element (k within 128-chunk, col), chunk `seg`:
//   kb=k>>2, j=k&1, kh=(k>>1)&1, n=col>>4, c=col&15, L=kh*16+c
//   word = seg*16384 + (kb*8 + n)*64 + L*2 + j
// ---------------------------------------------------------------------------
__device__ __forceinline__ void stage_w_async(const float* __restrict__ w0,
                                              float* lds, int tid, int rows) {
  const int total = rows * DF / 4;  // float4-granular global reads
  for (int it = tid; it < total; it += 256) {
    const int idx = it * 4;
    const int krow = idx >> 7;     // 0..rows-1
    const int col = idx & 127;     // multiple of 4
    const int seg = krow >> 7;
    const int k = krow & 127;
    const int kb = k >> 2, j = k & 1, kh = (k >> 1) & 1;
    const int n = col >> 4, cc = col & 15;
    const int word = seg * 16384 + (kb * 8 + n) * 64 + (kh * 16 + cc) * 2 + j;
    // low 32 bits of a generic LDS pointer == LDS byte offset (aperture rule)
    unsigned laddr = (unsigned)(uintptr_t)(lds + word);
    unsigned long long gaddr = (unsigned long long)(uintptr_t)(w0 + idx);
#pragma unroll
    for (int i = 0; i < 4; ++i) {
      asm volatile("global_load_async_to_lds_b32 %0, %1, off"
                   :: "v"(laddr + 8u * i), "v"(gaddr + 4ull * i)
                   : "memory");
    }
  }
  asm volatile("s_wait_asynccnt 0x0" ::: "memory");
}

// 32 k-blocks (K=128) of WMMA into 8 accumulators (16x128 C tile).
// A-fragment (ISA): lane L, row M=L&15; lanes 0-15 load K0,K0+1; 16-31 K0+2,K0+3.
__device__ __forceinline__ void mm_chunk(const float* __restrict__ arow,
                                         const float* lds, int lane, int half,
                                         v8f acc[8]) {
#pragma unroll 4
  for (int kb = 0; kb < 32; ++kb) {
    v2f a = *(const v2f*)(arow + kb * 4 + half * 2);
    const float* bb = lds + kb * 512 + lane * 2;
#pragma unroll
    for (int n = 0; n < 8; ++n) {
      v2f b = *(const v2f*)(bb + n * 64);
      acc[n] = __builtin_amdgcn_wmma_f32_16x16x4_f32(
          false, a, false, b, (short)0, acc[n], false, false);
    }
  }
}

__device__ __forceinline__ void mm_chunk_scaled(const float* __restrict__ arow,
                                                float scale, const float* lds,
                                                int lane, int half, v8f acc[8]) {
#pragma unroll 4
  for (int kb = 0; kb < 32; ++kb) {
    v2f a = *(const v2f*)(arow + kb * 4 + half * 2);
    a.x *= scale;
    a.y *= scale;
    const float* bb = lds + kb * 512 + lane * 2;
#pragma unroll
    for (int n = 0; n < 8; ++n) {
      v2f b = *(const v2f*)(bb + n * 64);
      acc[n] = __builtin_amdgcn_wmma_f32_16x16x4_f32(
          false, a, false, b, (short)0, acc[n], false, false);
    }
  }
}

// Residual-add + bias, LayerNorm over 128 cols per row (C layout: VGPR v of
// lane L holds row M = v + 8*(L>>4), col = 16n + (L&15)), then scale/shift/store.
__device__ __forceinline__ void finalize_ln(v8f acc[8],
                                            const float* __restrict__ resid,
                                            const float* __restrict__ bias,
                                            const float* __restrict__ gamma,
                                            const float* __restrict__ beta,
                                            float* __restrict__ out,
                                            size_t row0, int c, int half) {
  float s1[8], s2[8];
#pragma unroll
  for (int v = 0; v < 8; ++v) { s1[v] = 0.f; s2[v] = 0.f; }
#pragma unroll
  for (int n = 0; n < 8; ++n) {
    const int col = n * 16 + c;
    const float b = bias[col];
#pragma unroll
    for (int v = 0; v < 8; ++v) {
      const size_t row = row0 + v + 8 * half;
      float y = acc[n][v] + resid[row * DF + col] + b;
      acc[n][v] = y;
      s1[v] += y;
      s2[v] += y * y;
    }
  }
  // butterfly reduce within each 16-lane half (rows differ between halves)
#pragma unroll
  for (int v = 0; v < 8; ++v) {
#pragma unroll
    for (int mask = 1; mask < 16; mask <<= 1) {
      s1[v] += __shfl_xor(s1[v], mask, 32);
      s2[v] += __shfl_xor(s2[v], mask, 32);
    }
  }
  float mu[8], rs[8];
#pragma unroll
  for (int v = 0; v < 8; ++v) {
    mu[v] = s1[v] * (1.f / 128.f);
    float var = s2[v] * (1.f / 128.f) - mu[v] * mu[v];
    rs[v] = rsqrtf(var + LN_EPS);
  }
#pragma unroll
  for (int n = 0; n < 8; ++n) {
    const int col = n * 16 + c;
    const float g = gamma[col];
    const float bb = beta[col];
#pragma unroll
    for (int v = 0; v < 8; ++v) {
      const size_t row = row0 + v + 8 * half;
      out[row * DF + col] = (acc[n][v] - mu[v]) * rs[v] * g + bb;
    }
  }
}

// ---------------------------------------------------------------------------
// Kernel 1: edge update.  All of W_e (384x128 = 192 KB, fragment-swizzled)
// resident in dynamic LDS; staged once per WG via async-to-LDS, then each
// wave runs a persistent grid-stride loop over 16-edge tiles (no barriers in
// steady state).  K=384 = three 128-chunks matching concat([edge, src, tgt]).
// ---------------------------------------------------------------------------
__global__ __launch_bounds__(256) void edge_update_kernel(
    const float* __restrict__ nodef, const float* __restrict__ edgef,
    const int* __restrict__ eidx, const float* __restrict__ We,
    const float* __restrict__ be, const float* __restrict__ ge,
    const float* __restrict__ beln, float* __restrict__ out_edges) {
  extern __shared__ float lds[];  // 3 * 16384 floats = 192 KB
  const int tid = threadIdx.x;
  const int lane = tid & 31;
  const int wave = tid >> 5;
  const int c = lane & 15;
  const int half = lane >> 4;

  stage_w_async(We, lds, tid, 384);
  __syncthreads();

  const size_t stride = (size_t)gridDim.x * 8;
  for (size_t tile = (size_t)blockIdx.x * 8 + wave; tile < NE / 16;
       tile += stride) {
    const size_t e = tile * 16 + c;  // A row for this lane
    const int s = eidx[e];
    const int t = eidx[NE + e];
    v8f acc[8] = {};
    mm_chunk(edgef + e * (size_t)DF, lds, lane, half, acc);
    mm_chunk(nodef + (size_t)s * DF, lds + 16384, lane, half, acc);
    mm_chunk(nodef + (size_t)t * DF, lds + 32768, lane, half, acc);
    finalize_ln(acc, edgef, be, ge, beln, out_edges, tile * 16, c, half);
  }
}

// ---------------------------------------------------------------------------
// Kernel 2: zero accumulators, then scatter new_edges to both endpoints.
// One thread handles 4 columns of one edge (b128 load, 8 f32 atomics).
// ---------------------------------------------------------------------------
__global__ void zero_kernel(float4* __restrict__ p, size_t n4) {
  size_t i = (size_t)blockIdx.x * 256 + threadIdx.x;
  if (i < n4) p[i] = make_float4(0.f, 0.f, 0.f, 0.f);
}

__global__ void scatter_kernel(const float* __restrict__ new_edges,
                               const int* __restrict__ eidx,
                               float* __restrict__ accum,
                               float* __restrict__ cnt) {
  const size_t idx = (size_t)blockIdx.x * 256 + threadIdx.x;  // e*32 + q/4
  const size_t e = idx >> 5;
  const int q = (int)(idx & 31) * 4;
  const float4 v = *(const float4*)(new_edges + e * DF + q);
  const int s = eidx[e];
  const int t = eidx[NE + e];
  float* as = accum + (size_t)s * DF + q;
  float* at = accum + (size_t)t * DF + q;
  atomicAdd(as + 0, v.x);
  atomicAdd(as + 1, v.y);
  atomicAdd(as + 2, v.z);
  atomicAdd(as + 3, v.w);
  atomicAdd(at + 0, v.x);
  atomicAdd(at + 1, v.y);
  atomicAdd(at + 2, v.z);
  atomicAdd(at + 3, v.w);
  if (q == 0) {
    atomicAdd(cnt + s, 1.f);
    atomicAdd(cnt + t, 1.f);
  }
}

// ---------------------------------------------------------------------------
// Kernel 3: node update.  W_n (256x128 = 128 KB) fully LDS-resident;
// persistent tile loop.  K=256 = [node_features | edge_mean] chunks with
// edge_mean computed on the fly as accum * 1/(cnt + 1e-10).
// ---------------------------------------------------------------------------
__global__ __launch_bounds__(256) void node_update_kernel(
    const float* __restrict__ nodef, const float* __restrict__ Wn,
    const float* __restrict__ bn, const float* __restrict__ gn,
    const float* __restrict__ bnln, const float* __restrict__ accum,
    const float* __restrict__ cnt, float* __restrict__ out_nodes) {
  extern __shared__ float lds[];  // 2 * 16384 floats = 128 KB
  const int tid = threadIdx.x;
  const int lane = tid & 31;
  const int wave = tid >> 5;
  const int c = lane & 15;
  const int half = lane >> 4;

  stage_w_async(Wn, lds, tid, 256);
  __syncthreads();

  const size_t stride = (size_t)gridDim.x * 8;
  for (size_t tile = (size_t)blockIdx.x * 8 + wave; tile < NN / 16;
       tile += stride) {
    const size_t r = tile * 16 + c;
    const float inv = 1.f / (cnt[r] + 1e-10f);
    v8f acc[8] = {};
    mm_chunk(nodef + r * (size_t)DF, lds, lane, half, acc);
    mm_chunk_scaled(accum + r * (size_t)DF, inv, lds + 16384, lane, half, acc);
    finalize_ln(acc, nodef, bn, gn, bnln, out_nodes, tile * 16, c, half);
  }
}

// ---------------------------------------------------------------------------
extern "C" void kernel_launch(void* const* d_in, const int* in_sizes, int n_in,
                              void* d_out, int out_size, void* d_ws,
                              size_t ws_size, hipStream_t stream) {
  const float* nodef = (const float*)d_in[0];
  const float* edgef = (const float*)d_in[1];
  const int* eidx    = (const int*)d_in[2];
  const float* We    = (const float*)d_in[3];
  const float* be    = (const float*)d_in[4];
  const float* ge    = (const float*)d_in[5];
  const float* beln  = (const float*)d_in[6];
  const float* Wn    = (const float*)d_in[7];
  const float* bn    = (const float*)d_in[8];
  const float* gn    = (const float*)d_in[9];
  const float* bnln  = (const float*)d_in[10];

  float* out_nodes = (float*)d_out;                // [NN,128]
  float* out_edges = out_nodes + (size_t)NN * DF;  // [NE,128]
  float* accum = (float*)d_ws;                     // [NN,128]
  float* cnt   = accum + (size_t)NN * DF;          // [NN]

  const size_t zn4 = ((size_t)NN * DF + NN) / 4;  // divides exactly
  zero_kernel<<<(unsigned)((zn4 + 255) / 256), 256, 0, stream>>>(
      (float4*)accum, zn4);

  edge_update_kernel<<<512, 256, 384 * DF * sizeof(float), stream>>>(
      nodef, edgef, eidx, We, be, ge, beln, out_edges);

  scatter_kernel<<<(unsigned)(((size_t)NE * 32) / 256), 256, 0, stream>>>(
      out_edges, eidx, accum, cnt);

  node_update_kernel<<<128, 256, 256 * DF * sizeof(float), stream>>>(
      nodef, Wn, bn, gn, bnln, accum, cnt, out_nodes);
}